// GridSamplePScan_45509473468507
// MI455X (gfx1250) — compile-verified
//
#include <hip/hip_runtime.h>
#include <hip/hip_bf16.h>

// GridSamplePScan for MI455X (gfx1250, wave32).
// B=4, L=32, C=16, H=W=128, float32 in/out.
// Memory-bound gather/scan: no matrix contraction -> WMMA inapplicable.
// gfx1250-specific paths used: global_load_async_to_lds_b128 /
// global_store_async_from_lds_b128 + s_wait_asynccnt (prefix copies),
// global_prefetch_b8 (gather planes), wave32 tiling, B128 streaming.

#define PS_H   128
#define PS_W   128
#define PS_HW  (128 * 128)
#define PS_B   4
#define PS_L   32
#define PS_C   16

// ---------------------------------------------------------------------------
// Compute kernel: one block = (b, l, 8-row tile). 256 threads, 4 px/thread.
// new_f[l] = f[l] + bilinear(f[l-step], grid(f[l]), border)
// new_i[l] = i[l] + bilinear(i[l-step], grid(f[l]), zeros)
// ---------------------------------------------------------------------------
__global__ __launch_bounds__(256) void pscan_step_kernel(
    const float* __restrict__ fSrc, const float* __restrict__ iSrc,
    float* __restrict__ fDst, float* __restrict__ iDst,
    int step, int L)
{
    const int tile = blockIdx.x;            // 0..15 (8 rows each)
    const int l    = blockIdx.y + step;     // updated position
    const int b    = blockIdx.z;
    const int lp   = l - step;              // previous position
    const int tid  = (int)threadIdx.x;
    const int row  = tid >> 5;              // 0..7
    const int x0p  = (tid & 31) << 2;       // 0,4,...,124
    const int y    = (tile << 3) + row;     // 0..127
    const int pix  = (y << 7) + x0p;

    const size_t fC = (((size_t)b * L) + (size_t)l ) * (size_t)(2 * PS_HW);
    const size_t fP = (((size_t)b * L) + (size_t)lp) * (size_t)(2 * PS_HW);
    const size_t iC = (((size_t)b * L) + (size_t)l ) * (size_t)(PS_C * PS_HW);
    const size_t iP = (((size_t)b * L) + (size_t)lp) * (size_t)(PS_C * PS_HW);

    // current flow (grid source), vectorized B128 loads
    const float4 fxv = *(const float4*)(fSrc + fC + pix);
    const float4 fyv = *(const float4*)(fSrc + fC + PS_HW + pix);
    float fxa[4] = {fxv.x, fxv.y, fxv.z, fxv.w};
    float fya[4] = {fyv.x, fyv.y, fyv.z, fyv.w};

    int   o00[4], o01[4], o10[4], o11[4];       // clamped gather offsets
    float wb00[4], wb01[4], wb10[4], wb11[4];   // border weights (flows)
    float wz00[4], wz01[4], wz10[4], wz11[4];   // zero-masked weights (images)

#pragma unroll
    for (int k = 0; k < 4; ++k) {
        // gx+1 = (x+0.5)/64 + fx ; wrap x: remainder(.,2) ; unnormalize (AC=F)
        const float t  = ((float)(x0p + k) + 0.5f) * (1.0f / 64.0f) + fxa[k];
        const float r  = t - floorf(t * 0.5f) * 2.0f;      // in [0,2]
        const float px = r * 64.0f - 0.5f;
        const float py = (float)y + fya[k] * 64.0f;        // (gy+1)*64-0.5 simplified

        const float xf = floorf(px), yf = floorf(py);
        const float wx = px - xf,   wy = py - yf;
        const int ix0 = (int)xf, iy0 = (int)yf;
        const int ix1 = ix0 + 1, iy1 = iy0 + 1;
        const int xc0 = min(max(ix0, 0), PS_W - 1), xc1 = min(max(ix1, 0), PS_W - 1);
        const int yc0 = min(max(iy0, 0), PS_H - 1), yc1 = min(max(iy1, 0), PS_H - 1);
        o00[k] = (yc0 << 7) + xc0;  o01[k] = (yc0 << 7) + xc1;
        o10[k] = (yc1 << 7) + xc0;  o11[k] = (yc1 << 7) + xc1;

        const float a  = (1.0f - wx) * (1.0f - wy);
        const float bq = wx * (1.0f - wy);
        const float c  = (1.0f - wx) * wy;
        const float d  = wx * wy;
        wb00[k] = a; wb01[k] = bq; wb10[k] = c; wb11[k] = d;

        const float vx0 = (ix0 >= 0 && ix0 < PS_W) ? 1.0f : 0.0f;
        const float vx1 = (ix1 >= 0 && ix1 < PS_W) ? 1.0f : 0.0f;
        const float vy0 = (iy0 >= 0 && iy0 < PS_H) ? 1.0f : 0.0f;
        const float vy1 = (iy1 >= 0 && iy1 < PS_H) ? 1.0f : 0.0f;
        wz00[k] = a  * vy0 * vx0;  wz01[k] = bq * vy0 * vx1;
        wz10[k] = c  * vy1 * vx0;  wz11[k] = d  * vy1 * vx1;
    }

    // ---- flows: border padding ----
    {
        const float* P0 = fSrc + fP;
        const float* P1 = fSrc + fP + PS_HW;
        __builtin_prefetch(iSrc + iP + o00[0], 0, 0);   // warm first image plane
        float ofx[4], ofy[4];
#pragma unroll
        for (int k = 0; k < 4; ++k) {
            ofx[k] = fxa[k] + wb00[k] * P0[o00[k]] + wb01[k] * P0[o01[k]]
                             + wb10[k] * P0[o10[k]] + wb11[k] * P0[o11[k]];
            ofy[k] = fya[k] + wb00[k] * P1[o00[k]] + wb01[k] * P1[o01[k]]
                             + wb10[k] * P1[o10[k]] + wb11[k] * P1[o11[k]];
        }
        *(float4*)(fDst + fC + pix)         = make_float4(ofx[0], ofx[1], ofx[2], ofx[3]);
        *(float4*)(fDst + fC + PS_HW + pix) = make_float4(ofy[0], ofy[1], ofy[2], ofy[3]);
    }

    // ---- images: zeros padding, 16 channels, shared corner offsets ----
    for (int c = 0; c < PS_C; ++c) {
        const float* P = iSrc + iP + (size_t)c * PS_HW;
        if (c + 1 < PS_C)
            __builtin_prefetch(iSrc + iP + (size_t)(c + 1) * PS_HW + o00[0], 0, 0);
        const float4 icv = *(const float4*)(iSrc + iC + (size_t)c * PS_HW + pix);
        float ica[4] = {icv.x, icv.y, icv.z, icv.w};
        float outa[4];
#pragma unroll
        for (int k = 0; k < 4; ++k) {
            outa[k] = ica[k] + wz00[k] * P[o00[k]] + wz01[k] * P[o01[k]]
                              + wz10[k] * P[o10[k]] + wz11[k] * P[o11[k]];
        }
        *(float4*)(iDst + iC + (size_t)c * PS_HW + pix) =
            make_float4(outa[0], outa[1], outa[2], outa[3]);
    }
}

// ---------------------------------------------------------------------------
// Prefix copy kernel: positions l < step are passed through unchanged.
// Pure data movement -> gfx1250 async Global->LDS->Global path, no VGPR
// round trip. One block = (b, l, 8-row tile); 18 planes (2 flow + 16 image)
// staged through 9 LDS slots x 2 passes (36 KB static LDS).
// ---------------------------------------------------------------------------
__global__ __launch_bounds__(256) void pscan_copy_async_kernel(
    const float* __restrict__ fSrc, const float* __restrict__ iSrc,
    float* __restrict__ fDst, float* __restrict__ iDst, int L)
{
    __shared__ __align__(16) char smem[9 * 4096];

    const int tile = blockIdx.x;
    const int l    = blockIdx.y;
    const int b    = blockIdx.z;
    const int tid  = (int)threadIdx.x;
    const int row  = tid >> 5;
    const int xb   = (tid & 31) << 4;                  // 16B chunk within row
    const int off  = (((tile << 3) + row) << 9) + xb;  // byte offset in 64KB plane

    const unsigned ldsBase = (unsigned)(size_t)(void*)&smem[0]; // low 32b = LDS offset
    const size_t fB = (((size_t)b * L) + (size_t)l) * (size_t)(2 * PS_HW) * 4u;
    const size_t iB = (((size_t)b * L) + (size_t)l) * (size_t)(PS_C * PS_HW) * 4u;

#pragma unroll
    for (int g = 0; g < 2; ++g) {
#pragma unroll
        for (int j = 0; j < 9; ++j) {
            const int ch = g * 9 + j;
            const char* sp = (ch < 2)
                ? ((const char*)fSrc + fB + (size_t)ch * (PS_HW * 4))
                : ((const char*)iSrc + iB + (size_t)(ch - 2) * (PS_HW * 4));
            const unsigned lds = ldsBase + (unsigned)j * 4096u + (unsigned)tid * 16u;
            asm volatile("global_load_async_to_lds_b128 %0, %1, %2"
                         :: "v"(lds), "v"(off), "s"(sp) : "memory");
        }
        asm volatile("s_wait_asynccnt 0x0" ::: "memory");
#pragma unroll
        for (int j = 0; j < 9; ++j) {
            const int ch = g * 9 + j;
            char* dp = (ch < 2)
                ? ((char*)fDst + fB + (size_t)ch * (PS_HW * 4))
                : ((char*)iDst + iB + (size_t)(ch - 2) * (PS_HW * 4));
            const unsigned lds = ldsBase + (unsigned)j * 4096u + (unsigned)tid * 16u;
            asm volatile("global_store_async_from_lds_b128 %0, %1, %2"
                         :: "v"(off), "v"(lds), "s"(dp) : "memory");
        }
        asm volatile("s_wait_asynccnt 0x0" ::: "memory");  // LDS slots reused next pass
    }
}

// ---------------------------------------------------------------------------
// Host: 5 scan rounds, each = compute (l>=step) + async prefix copy (l<step).
// Ping-pong: image buffer A = d_out (odd #rounds -> final lands in d_out),
// image buffer B + two flow buffers in d_ws (needs 160 MB scratch).
// ---------------------------------------------------------------------------
extern "C" void kernel_launch(void* const* d_in, const int* in_sizes, int n_in,
                              void* d_out, int out_size, void* d_ws, size_t ws_size,
                              hipStream_t stream)
{
    (void)in_sizes; (void)n_in; (void)out_size;

    const int L = PS_L;
    const float* flows_in  = (const float*)d_in[0];
    const float* images_in = (const float*)d_in[1];
    float* out = (float*)d_out;

    const size_t fElems = (size_t)PS_B * L * 2    * PS_HW;   // 4M floats  (16 MB)
    const size_t iElems = (size_t)PS_B * L * PS_C * PS_HW;   // 32M floats (128 MB)
    (void)iElems; (void)ws_size;                             // need >= 160 MB scratch

    float* fBufA = (float*)d_ws;
    float* fBufB = fBufA + fElems;
    float* iBufB = fBufB + fElems;
    float* iBufA = out;

    dim3 block(256);
    const float* fs = flows_in;
    const float* is = images_in;
    int parity = 0;
    for (int s = 1; s < L; s <<= 1) {
        float* fd = (parity == 0) ? fBufA : fBufB;
        float* id = (parity == 0) ? iBufA : iBufB;

        dim3 gridC(16, L - s, PS_B);
        pscan_step_kernel<<<gridC, block, 0, stream>>>(fs, is, fd, id, s, L);

        dim3 gridP(16, s, PS_B);
        pscan_copy_async_kernel<<<gridP, block, 0, stream>>>(fs, is, fd, id, L);

        fs = fd; is = id;
        parity ^= 1;
    }
}